// ContactMapLoss_47519518163566
// MI455X (gfx1250) — compile-verified
//
#include <hip/hip_runtime.h>

// Problem constants (from reference)
#define BB 8
#define VV 10475
#define RR 75
#define MM 40
#define NPAIR (BB * RR * RR)      // 45000
#define NPTS  (BB * RR * MM)      // 24000

typedef float v2f __attribute__((ext_vector_type(2)));
typedef float v8f __attribute__((ext_vector_type(8)));

#define BIGF 3.0e38f

// ---------------------------------------------------------------------------
// Kernel 1: gather points via rid_to_vid, pad to (x,y,z,0), compute |p|^2
// ---------------------------------------------------------------------------
__global__ __launch_bounds__(256)
void cml_gather_kernel(const float* __restrict__ v1,
                       const float* __restrict__ v2,
                       const long long* __restrict__ rid_to_vid,
                       float4* __restrict__ p1g, float4* __restrict__ p2g,
                       float* __restrict__ n1g, float* __restrict__ n2g)
{
    int t = blockIdx.x * blockDim.x + threadIdx.x;
    if (t >= 2 * NPTS) return;
    int which = (t >= NPTS) ? 1 : 0;
    int idx = which ? (t - NPTS) : t;          // [b, r, m] flattened
    int rm = idx % (RR * MM);                  // r*M + m  (same for every b)
    long long vid = rid_to_vid[rm];
    int b = idx / (RR * MM);
    const float* src = which ? v2 : v1;
    const float* p = src + ((long long)b * VV + vid) * 3;
    float x = p[0], y = p[1], z = p[2];
    float n = x * x + y * y + z * z;
    float4 q = make_float4(x, y, z, 0.0f);
    if (which) { p2g[idx] = q; n2g[idx] = n; }
    else       { p1g[idx] = q; n1g[idx] = n; }
}

// ---------------------------------------------------------------------------
// Kernel 2: one wave32 per (b,r,s) pair. 3x3 tiles of 16x16 dots via
// V_WMMA_F32_16X16X4_F32, epilogue in VALU, min-reductions via shfl_xor.
// ---------------------------------------------------------------------------
__global__ __launch_bounds__(256)
void cml_pair_kernel(const float4* __restrict__ p1g,
                     const float4* __restrict__ p2g,
                     const float* __restrict__ n1g,
                     const float* __restrict__ n2g,
                     const int* __restrict__ cmap,
                     float* __restrict__ pair_loss)
{
    const int lane = threadIdx.x & 31;
    const int wave = threadIdx.x >> 5;
    const int pair = blockIdx.x * 8 + wave;           // 8 waves / block
    if (pair >= NPAIR) return;                        // wave-uniform

    const int b  = pair / (RR * RR);
    const int rs = pair % (RR * RR);
    const int r  = rs / RR;
    const int s  = rs % RR;

    float pl = 0.0f;
    if (cmap[pair] != 0) {                            // wave-uniform branch
        const int half = lane >> 4;                   // 0: K=0,1  1: K=2,3
        const int ln   = lane & 15;

        const int base1 = (b * RR + r) * MM;
        const int base2 = (b * RR + s) * MM;

        // --- B fragments (4x16, K in VGPR, N = ln) + per-column n2 ---
        v2f   Bfrag[3];
        float n2v[3];
        bool  validJ[3];
#pragma unroll
        for (int tj = 0; tj < 3; ++tj) {
            int Nj = tj * 16 + ln;
            validJ[tj] = (Nj < MM);
            int idx = base2 + (validJ[tj] ? Nj : (MM - 1));
            float4 q = p2g[idx];
            v2f f;
            if (half) { f.x = q.z; f.y = 0.0f; }
            else      { f.x = q.x; f.y = q.y;  }
            Bfrag[tj] = f;
            n2v[tj]   = n2g[idx];
        }

        float colmin[3] = {BIGF, BIGF, BIGF};
        float rowsum = 0.0f;

#pragma unroll
        for (int ti = 0; ti < 3; ++ti) {
            // --- A fragment (16x4): row M = ti*16+ln, K split across halves ---
            int Mi = ti * 16 + ln;
            int idxA = base1 + ((Mi < MM) ? Mi : (MM - 1));
            float4 p = p1g[idxA];
            v2f Afrag;
            if (half) { Afrag.x = p.z; Afrag.y = 0.0f; }
            else      { Afrag.x = p.x; Afrag.y = p.y;  }

            // n1 for the 8 rows this lane's accumulator holds
            float n1v[8];
            bool  validI[8];
#pragma unroll
            for (int k = 0; k < 8; ++k) {
                int Mi2 = ti * 16 + k + 8 * half;
                validI[k] = (Mi2 < MM);
                n1v[k] = n1g[base1 + (validI[k] ? Mi2 : (MM - 1))];
            }

            float rowmin[8];
#pragma unroll
            for (int k = 0; k < 8; ++k) rowmin[k] = BIGF;

#pragma unroll
            for (int tj = 0; tj < 3; ++tj) {
                v8f acc = {};
                acc = __builtin_amdgcn_wmma_f32_16x16x4_f32(
                    /*neg_a=*/false, Afrag, /*neg_b=*/false, Bfrag[tj],
                    /*c_mod=*/(short)0, acc, /*reuse_a=*/false, /*reuse_b=*/false);

                float cmin = BIGF;
#pragma unroll
                for (int k = 0; k < 8; ++k) {
                    float d2 = fmaxf(n1v[k] + n2v[tj] - 2.0f * acc[k], 0.0f);
                    cmin      = fminf(cmin, validI[k] ? d2 : BIGF);
                    rowmin[k] = fminf(rowmin[k], validJ[tj] ? d2 : BIGF);
                }
                // combine the two lane-halves (rows k vs k+8) for this column
                cmin = fminf(cmin, __shfl_xor(cmin, 16, 32));
                colmin[tj] = fminf(colmin[tj], cmin);
            }

            // min over the 16 columns within each lane-half, per row k
#pragma unroll
            for (int k = 0; k < 8; ++k) {
                float rm = rowmin[k];
                rm = fminf(rm, __shfl_xor(rm, 1, 32));
                rm = fminf(rm, __shfl_xor(rm, 2, 32));
                rm = fminf(rm, __shfl_xor(rm, 4, 32));
                rm = fminf(rm, __shfl_xor(rm, 8, 32));
                if (ln == 0 && validI[k]) rowsum += rm;  // once per half
            }
        }

        // column-min contribution: lanes 0..15 own columns tj*16+ln
        float total = rowsum;
#pragma unroll
        for (int tj = 0; tj < 3; ++tj)
            if (half == 0 && validJ[tj]) total += colmin[tj];

        // wave-wide sum
        total += __shfl_xor(total, 16, 32);
        total += __shfl_xor(total, 8, 32);
        total += __shfl_xor(total, 4, 32);
        total += __shfl_xor(total, 2, 32);
        total += __shfl_xor(total, 1, 32);
        pl = total * (1.0f / (float)MM);
    }

    if (lane == 0) pair_loss[pair] = pl;
}

// ---------------------------------------------------------------------------
// Kernel 3: deterministic per-batch reduction of pair_loss -> distance[b]
// ---------------------------------------------------------------------------
__global__ __launch_bounds__(256)
void cml_reduce_kernel(const float* __restrict__ pair_loss,
                       float* __restrict__ out)
{
    __shared__ float sm[256];
    const int b = blockIdx.x;
    const float* pl = pair_loss + b * RR * RR;
    float acc = 0.0f;
    for (int i = threadIdx.x; i < RR * RR; i += 256) acc += pl[i];
    sm[threadIdx.x] = acc;
    __syncthreads();
    for (int st = 128; st > 0; st >>= 1) {
        if ((int)threadIdx.x < st) sm[threadIdx.x] += sm[threadIdx.x + st];
        __syncthreads();
    }
    if (threadIdx.x == 0) out[b] = sm[0];
}

// ---------------------------------------------------------------------------
extern "C" void kernel_launch(void* const* d_in, const int* in_sizes, int n_in,
                              void* d_out, int out_size, void* d_ws, size_t ws_size,
                              hipStream_t stream)
{
    const float*     v1   = (const float*)d_in[0];
    const float*     v2   = (const float*)d_in[1];
    const int*       cmap = (const int*)d_in[2];
    const long long* r2v  = (const long long*)d_in[3];

    float* ws = (float*)d_ws;
    float4* p1g      = (float4*)(ws);                  //  96000 floats
    float4* p2g      = (float4*)(ws + 96000);          //  96000 floats
    float*  n1g      = ws + 192000;                    //  24000 floats
    float*  n2g      = ws + 216000;                    //  24000 floats
    float*  ploss    = ws + 240000;                    //  45000 floats

    // 1) gather + norms: 2*24000 threads
    {
        int threads = 2 * NPTS;
        int blocks = (threads + 255) / 256;
        cml_gather_kernel<<<blocks, 256, 0, stream>>>(v1, v2, r2v, p1g, p2g, n1g, n2g);
    }
    // 2) one wave32 per (b,r,s): 45000 waves, 8 waves per block
    {
        int blocks = (NPAIR + 7) / 8;   // 5625
        cml_pair_kernel<<<blocks, 256, 0, stream>>>(p1g, p2g, n1g, n2g, cmap, ploss);
    }
    // 3) per-batch reduction
    cml_reduce_kernel<<<BB, 256, 0, stream>>>(ploss, (float*)d_out);
}